// GIN_88871463289129
// MI455X (gfx1250) — compile-verified
//
#include <hip/hip_runtime.h>
#include <hip/hip_bf16.h>
#include <math.h>

// ---------------------------------------------------------------------------
// GIN forward on MI455X (gfx1250, wave32).
// Dense layers use V_WMMA_F32_16X16X4_F32 (full-precision fp32 WMMA).
// Scatter-add / pooling go through global_atomic_add_f32 (L2-resident: h and
// agg are 20.5 MB each << 192 MB L2).
// ---------------------------------------------------------------------------

#define N_NODES   40000
#define N_EDGES   640000
#define D_HID     128
#define N_GRAPHS  64
#define N_CLASSES 10
#define BN_EPS    1e-5f

typedef __attribute__((ext_vector_type(2))) float v2f;
typedef __attribute__((ext_vector_type(8))) float v8f;

__device__ __forceinline__ float elu1(float x) {
    return x > 0.0f ? x : (__expf(x) - 1.0f);
}

// ---------------------------------------------------------------------------
// Fused GEMM: C[M,128] = f_pre(A)[M,128] @ W[128,128] + bias, with epilogue.
//   PRE : 0 = identity            (A)
//         1 = A + A2              (h + agg for GIN conv)
//         2 = elu(bn(A))          (BatchNorm+ELU fused into A load)
//   POST: 0 = bias only
//         1 = bias + accumulate per-column sum/sumsq into stats (BN train stats)
//         2 = bias + outer ELU
// Block = 256 threads = 8 waves. Wave w computes rows [blk*128+16w, +16) x all
// 128 cols as 8 WMMA 16x16 tiles, K-loop step 4 (V_WMMA_F32_16X16X4_F32).
// W is staged once into LDS (64 KB).
// ---------------------------------------------------------------------------
template <int PRE, int POST>
__global__ __launch_bounds__(256)
void gemm128_wmma(const float* __restrict__ A, const float* __restrict__ A2,
                  const float* __restrict__ W, const float* __restrict__ bias,
                  const float* __restrict__ mu, const float* __restrict__ rsig,
                  const float* __restrict__ gamma, const float* __restrict__ beta,
                  float* __restrict__ C, float* __restrict__ stats, int M)
{
    __shared__ float sW[D_HID * D_HID];  // 64 KB: full weight matrix, row-major [K][N]

    // Cooperative 128-bit loads of W into LDS.
    for (int i = threadIdx.x; i < (D_HID * D_HID) / 4; i += 256) {
        ((float4*)sW)[i] = ((const float4*)W)[i];
    }
    __syncthreads();

    const int lane = threadIdx.x & 31;
    const int wave = threadIdx.x >> 5;
    const int row0 = blockIdx.x * 128 + wave * 16;  // 16-row strip per wave
    const int mrow = lane & 15;                     // M (for A) / N (for B,C) sub-index
    const int koff = (lane >> 4) * 2;               // K sub-offset: lanes 16-31 handle K+2,K+3

    v8f acc[8] = {};  // 8 col-tiles x 8 rows each = 16x128 output strip

    // Clamp row for tail block: extra rows compute garbage but are never stored.
    int m = row0 + mrow;
    if (m > M - 1) m = M - 1;
    const float* Arow  = A + (size_t)m * D_HID;
    const float* A2row = (PRE == 1) ? (A2 + (size_t)m * D_HID) : nullptr;

    // K loop: uniform control flow, EXEC all-ones through every WMMA.
    for (int k = 0; k < D_HID; k += 4) {
        const int ka = k + koff;
        // A-frag (ISA 32-bit A 16x4 layout): lane holds A[m][ka], A[m][ka+1].
        v2f a;
        a.x = Arow[ka];
        a.y = Arow[ka + 1];
        if (PRE == 1) {
            a.x += A2row[ka];
            a.y += A2row[ka + 1];
        }
        if (PRE == 2) {
            float x0 = (a.x - mu[ka])     * rsig[ka]     * gamma[ka]     + beta[ka];
            float x1 = (a.y - mu[ka + 1]) * rsig[ka + 1] * gamma[ka + 1] + beta[ka + 1];
            a.x = elu1(x0);
            a.y = elu1(x1);
        }
#pragma unroll
        for (int t = 0; t < 8; ++t) {
            const int n = t * 16 + mrow;
            // B-frag (4x16, K rows striped across lanes): lane holds W[ka][n], W[ka+1][n].
            v2f b;
            b.x = sW[ka * D_HID + n];
            b.y = sW[(ka + 1) * D_HID + n];
            acc[t] = __builtin_amdgcn_wmma_f32_16x16x4_f32(
                /*neg_a=*/false, a, /*neg_b=*/false, b,
                /*c_mod=*/(short)0, acc[t], /*reuse_a=*/false, /*reuse_b=*/false);
        }
    }

    if (row0 >= M) return;  // wave-uniform: whole strip out of range

    // C layout: VGPR r -> row row0 + r + (lane<16 ? 0 : 8), col n = t*16 + (lane&15).
    const int rbase = row0 + ((lane >> 4) << 3);
#pragma unroll
    for (int t = 0; t < 8; ++t) {
        const int n = t * 16 + mrow;
        const float bi = bias[n];
        float s = 0.0f, sq = 0.0f;
#pragma unroll
        for (int r = 0; r < 8; ++r) {
            const int mm = rbase + r;
            if (mm < M) {
                float v = acc[t][r] + bi;
                if (POST == 2) v = elu1(v);
                C[(size_t)mm * D_HID + n] = v;
                if (POST == 1) { s += v; sq += v * v; }
            }
        }
        if (POST == 1) {
            atomicAdd(&stats[n], s);
            atomicAdd(&stats[D_HID + n], sq);
        }
    }
}

// ---------------------------------------------------------------------------
// Edge scatter: agg[dst[e]] += h[src[e]].  One edge per 32 threads; each
// thread moves a float4 (4 global_atomic_add_f32).
// ---------------------------------------------------------------------------
__global__ __launch_bounds__(256)
void scatter_add(const float* __restrict__ h, const int* __restrict__ src,
                 const int* __restrict__ dst, float* __restrict__ agg, int nEdges)
{
    const int t = blockIdx.x * 256 + threadIdx.x;
    const int e = t >> 5;
    const int lane = t & 31;
    if (e >= nEdges) return;
    const int s = src[e];
    const int d = dst[e];
    const float4 v = ((const float4*)(h + (size_t)s * D_HID))[lane];
    float* a = agg + (size_t)d * D_HID + lane * 4;
    atomicAdd(a + 0, v.x);
    atomicAdd(a + 1, v.y);
    atomicAdd(a + 2, v.z);
    atomicAdd(a + 3, v.w);
}

// ---------------------------------------------------------------------------
// BN train-mode stats finalize: mu, rsqrt(var + eps) from sum/sumsq.
// ---------------------------------------------------------------------------
__global__ void bn_finalize(const float* __restrict__ stats,
                            float* __restrict__ mu, float* __restrict__ rsig,
                            float invM)
{
    const int n = threadIdx.x;
    const float m = stats[n] * invM;
    float var = stats[D_HID + n] * invM - m * m;
    if (var < 0.0f) var = 0.0f;
    mu[n]   = m;
    rsig[n] = rsqrtf(var + BN_EPS);
}

// ---------------------------------------------------------------------------
// Global mean pool (accumulation): psum[batch[i]] += h[i]; pcnt[batch[i]] += 1.
// ---------------------------------------------------------------------------
__global__ __launch_bounds__(256)
void pool_sum(const float* __restrict__ h, const int* __restrict__ batch,
              float* __restrict__ psum, float* __restrict__ pcnt, int nNodes)
{
    const int t = blockIdx.x * 256 + threadIdx.x;
    const int node = t >> 5;
    const int lane = t & 31;
    if (node >= nNodes) return;
    const int g = batch[node];
    const float4 v = ((const float4*)(h + (size_t)node * D_HID))[lane];
    float* p = psum + (size_t)g * D_HID + lane * 4;
    atomicAdd(p + 0, v.x);
    atomicAdd(p + 1, v.y);
    atomicAdd(p + 2, v.z);
    atomicAdd(p + 3, v.w);
    if (lane == 0) atomicAdd(&pcnt[g], 1.0f);
}

// ---------------------------------------------------------------------------
// Final projection: out[g,c] = (psum[g]/max(cnt,1)) @ W_out[:,c] + b_out[c].
// 640 outputs, each a 128-length dot product — negligible.
// ---------------------------------------------------------------------------
__global__ __launch_bounds__(256)
void out_proj(const float* __restrict__ psum, const float* __restrict__ pcnt,
              const float* __restrict__ Wout, const float* __restrict__ bout,
              float* __restrict__ out)
{
    const int t = blockIdx.x * 256 + threadIdx.x;
    if (t >= N_GRAPHS * N_CLASSES) return;
    const int g = t / N_CLASSES;
    const int c = t % N_CLASSES;
    const float inv = 1.0f / fmaxf(pcnt[g], 1.0f);
    float acc = bout[c];
    for (int k = 0; k < D_HID; ++k)
        acc += psum[(size_t)g * D_HID + k] * inv * Wout[k * N_CLASSES + c];
    out[t] = acc;
}

// ---------------------------------------------------------------------------
// Host-side orchestration. All launches on `stream`; scratch in d_ws only.
// ---------------------------------------------------------------------------
extern "C" void kernel_launch(void* const* d_in, const int* in_sizes, int n_in,
                              void* d_out, int out_size, void* d_ws, size_t ws_size,
                              hipStream_t stream)
{
    (void)in_sizes; (void)n_in; (void)out_size; (void)ws_size;

    const float* x     = (const float*)d_in[0];
    const int*   ei    = (const int*)d_in[1];
    const int*   batch = (const int*)d_in[2];
    const float* W_in  = (const float*)d_in[3];
    const float* b_in  = (const float*)d_in[4];
    const float* cW1   = (const float*)d_in[5];
    const float* cb1   = (const float*)d_in[6];
    const float* cg    = (const float*)d_in[7];
    const float* cbe   = (const float*)d_in[8];
    const float* cW2   = (const float*)d_in[9];
    const float* cb2   = (const float*)d_in[10];
    const float* Wout  = (const float*)d_in[11];
    const float* bout  = (const float*)d_in[12];
    float* out = (float*)d_out;

    const int* src = ei;            // edge_index[0]
    const int* dst = ei + N_EDGES;  // edge_index[1]

    const size_t NM = (size_t)N_NODES * D_HID;  // 5.12M floats

    // Workspace layout (floats): h | agg | z | stats(256) | mu(128) | rsig(128)
    //                            | psum(64*128) | pcnt(64)
    float* h     = (float*)d_ws;
    float* agg   = h + NM;
    float* z     = agg + NM;
    float* stats = z + NM;
    float* mu    = stats + 2 * D_HID;
    float* rsig  = mu + D_HID;
    float* psum  = rsig + D_HID;
    float* pcnt  = psum + (size_t)N_GRAPHS * D_HID;

    const int gemmGrid    = (N_NODES + 127) / 128;        // 313
    const int scatterGrid = (N_EDGES * 32) / 256;         // 80000
    const int poolGrid    = (N_NODES * 32 + 255) / 256;   // 5000

    // 1) h = x @ W_in + b_in
    gemm128_wmma<0, 0><<<gemmGrid, 256, 0, stream>>>(
        x, nullptr, W_in, b_in, nullptr, nullptr, nullptr, nullptr,
        h, nullptr, N_NODES);

    // 2) Two GIN conv layers
    for (int l = 0; l < 2; ++l) {
        hipMemsetAsync(agg, 0, NM * sizeof(float), stream);
        hipMemsetAsync(stats, 0, 2 * D_HID * sizeof(float), stream);

        // agg = segment_sum(h[src], dst)
        scatter_add<<<scatterGrid, 256, 0, stream>>>(h, src, dst, agg, N_EDGES);

        // z = (h + agg) @ W1[l] + b1[l];  stats += column sums / sumsq
        gemm128_wmma<1, 1><<<gemmGrid, 256, 0, stream>>>(
            h, agg, cW1 + (size_t)l * D_HID * D_HID, cb1 + l * D_HID,
            nullptr, nullptr, nullptr, nullptr, z, stats, N_NODES);

        // mu, rsig
        bn_finalize<<<1, D_HID, 0, stream>>>(stats, mu, rsig, 1.0f / (float)N_NODES);

        // h = elu( elu(bn(z)) @ W2[l] + b2[l] )
        gemm128_wmma<2, 2><<<gemmGrid, 256, 0, stream>>>(
            z, nullptr, cW2 + (size_t)l * D_HID * D_HID, cb2 + l * D_HID,
            mu, rsig, cg + l * D_HID, cbe + l * D_HID, h, nullptr, N_NODES);
    }

    // 3) Global mean pool + out projection
    hipMemsetAsync(psum, 0, (size_t)N_GRAPHS * D_HID * sizeof(float), stream);
    hipMemsetAsync(pcnt, 0, N_GRAPHS * sizeof(float), stream);
    pool_sum<<<poolGrid, 256, 0, stream>>>(h, batch, psum, pcnt, N_NODES);
    out_proj<<<(N_GRAPHS * N_CLASSES + 255) / 256, 256, 0, stream>>>(
        psum, pcnt, Wout, bout, out);
}